// RGCN_20401094656588
// MI455X (gfx1250) — compile-verified
//
#include <hip/hip_runtime.h>

typedef float v2f __attribute__((ext_vector_type(2)));
typedef float v8f __attribute__((ext_vector_type(8)));

#define N_NODES 100000
#define E_EDGES 400000
#define R_REL   3
#define D_FEAT  128
#define C_CLS   40

// ---------------- degree / norm ----------------

__global__ __launch_bounds__(256)
void rgcn_deg_kernel(const int* __restrict__ edges,
                     float* __restrict__ deg_src,
                     float* __restrict__ deg_dst) {
    int idx = blockIdx.x * blockDim.x + threadIdx.x;   // over R*E
    if (idx >= R_REL * E_EDGES) return;
    int r = idx / E_EDGES;
    int e = idx - r * E_EDGES;
    int s = edges[(size_t)r * 2 * E_EDGES + e];
    int d = edges[(size_t)r * 2 * E_EDGES + E_EDGES + e];
    atomicAdd(&deg_src[r * N_NODES + s], 1.0f);
    atomicAdd(&deg_dst[r * N_NODES + d], 1.0f);
}

__global__ __launch_bounds__(256)
void rgcn_norm_kernel(float* __restrict__ deg, int n) {
    int i = blockIdx.x * blockDim.x + threadIdx.x;
    if (i < n) deg[i] = rsqrtf(fmaxf(deg[i], 1.0f));
}

// ---------------- WMMA GEMM: Y[Nx128] = (X .* rnorm[:,None]) @ W[128x128] ----------------
// Block: 256 threads = 8 waves. Each block: 16 rows. Each wave: one 16x16 col tile.
// fp32 WMMA 16x16x4: A 16x4 (lane<16: M=lane, K=k0..k0+1; lane>=16: M=lane-16, K=k0+2..k0+3)
//                    B 4x16 (lane<16: N=lane,  K=k0..k0+1; lane>=16: N=lane-16, K=k0+2..k0+3)
//                    C/D 16x16 (VGPR v: M=v+8*half, N=lane&15)

__global__ __launch_bounds__(256)
void rgcn_gemm128_wmma(const float* __restrict__ X,
                       const float* __restrict__ rnorm,
                       const float* __restrict__ W,
                       float* __restrict__ Y) {
    __shared__ float As[16][D_FEAT + 4];
    const int r0 = blockIdx.x * 16;

    for (int i = threadIdx.x; i < 16 * D_FEAT; i += 256) {
        int row = i >> 7, col = i & 127;
        As[row][col] = X[(size_t)(r0 + row) * D_FEAT + col] * rnorm[r0 + row];
    }
    __syncthreads();

    const int wave = threadIdx.x >> 5;     // 0..7 -> col tile
    const int lane = threadIdx.x & 31;
    const int half = lane >> 4;            // 0/1
    const int idx  = lane & 15;
    const int col0 = wave * 16;

    v8f c = {};
    #pragma unroll 8
    for (int k0 = 0; k0 < D_FEAT; k0 += 4) {
        v2f a, b;
        int k = k0 + 2 * half;
        a.x = As[idx][k + 0];
        a.y = As[idx][k + 1];
        b.x = W[(size_t)(k + 0) * D_FEAT + col0 + idx];
        b.y = W[(size_t)(k + 1) * D_FEAT + col0 + idx];
        c = __builtin_amdgcn_wmma_f32_16x16x4_f32(false, a, false, b,
                                                  (short)0, c, false, false);
    }

    #pragma unroll
    for (int v = 0; v < 8; ++v) {
        int row = v + 8 * half;
        Y[(size_t)(r0 + row) * D_FEAT + col0 + idx] = c[v];
    }
}

// ---------------- fused gather + dst-norm + scatter-add ----------------
// 32 threads per edge, float4 per thread (coalesced 512B row per edge group).

__global__ __launch_bounds__(256)
void rgcn_scatter_kernel(const float* __restrict__ hw,
                         const int* __restrict__ src,
                         const int* __restrict__ dst,
                         const float* __restrict__ dnorm,
                         float* __restrict__ hacc) {
    unsigned t = blockIdx.x * blockDim.x + threadIdx.x;  // E*32 threads
    unsigned e = t >> 5;
    if (e >= E_EDGES) return;
    int c4 = (int)(t & 31u) << 2;
    int s = src[e], d = dst[e];
    float dn = dnorm[d];
    const float4 v = *(const float4*)(hw + (size_t)s * D_FEAT + c4);
    float* o = hacc + (size_t)d * D_FEAT + c4;
    atomicAdd(o + 0, v.x * dn);
    atomicAdd(o + 1, v.y * dn);
    atomicAdd(o + 2, v.z * dn);
    atomicAdd(o + 3, v.w * dn);
}

// ---------------- finalize: h = [relu]((h + sum_r b_r) / R) ----------------

__global__ __launch_bounds__(256)
void rgcn_finalize_kernel(float* __restrict__ h,
                          const float* __restrict__ b,   // [R, D_FEAT]
                          int do_relu) {
    int t = blockIdx.x * blockDim.x + threadIdx.x;  // N*128 = 12.8M
    if (t >= N_NODES * D_FEAT) return;
    int f = t & 127;
    float bs = b[f] + b[D_FEAT + f] + b[2 * D_FEAT + f];
    float v = (h[t] + bs) * (1.0f / 3.0f);
    if (do_relu) v = fmaxf(v, 0.0f);
    h[t] = v;
}

// ---------------- classifier: out[Nx40] = H @ Wc[128x40] + bc ----------------
// Block: 96 threads = 3 waves (col tiles 0..2, tile 2 masked to cols 32..39).

__global__ __launch_bounds__(96)
void rgcn_cls_wmma(const float* __restrict__ H,
                   const float* __restrict__ Wc,
                   const float* __restrict__ bc,
                   float* __restrict__ out) {
    __shared__ float As[16][D_FEAT + 4];
    const int r0 = blockIdx.x * 16;

    for (int i = threadIdx.x; i < 16 * D_FEAT; i += 96) {
        int row = i >> 7, col = i & 127;
        As[row][col] = H[(size_t)(r0 + row) * D_FEAT + col];
    }
    __syncthreads();

    const int wave = threadIdx.x >> 5;
    const int lane = threadIdx.x & 31;
    const int half = lane >> 4;
    const int idx  = lane & 15;
    const int col  = wave * 16 + idx;
    const float msk = (col < C_CLS) ? 1.0f : 0.0f;
    const int   ccol = (col < C_CLS) ? col : (C_CLS - 1);

    v8f c = {};
    #pragma unroll 8
    for (int k0 = 0; k0 < D_FEAT; k0 += 4) {
        v2f a, b;
        int k = k0 + 2 * half;
        a.x = As[idx][k + 0];
        a.y = As[idx][k + 1];
        b.x = Wc[(size_t)(k + 0) * C_CLS + ccol] * msk;
        b.y = Wc[(size_t)(k + 1) * C_CLS + ccol] * msk;
        c = __builtin_amdgcn_wmma_f32_16x16x4_f32(false, a, false, b,
                                                  (short)0, c, false, false);
    }

    if (col < C_CLS) {
        float bv = bc[col];
        #pragma unroll
        for (int v = 0; v < 8; ++v) {
            int row = v + 8 * half;
            out[(size_t)(r0 + row) * C_CLS + col] = c[v] + bv;
        }
    }
}

// ---------------- launch ----------------

extern "C" void kernel_launch(void* const* d_in, const int* in_sizes, int n_in,
                              void* d_out, int out_size, void* d_ws, size_t ws_size,
                              hipStream_t stream) {
    (void)in_sizes; (void)n_in; (void)out_size; (void)ws_size;

    const float* x     = (const float*)d_in[0];
    const int*   edges = (const int*)  d_in[1];
    const float* W1    = (const float*)d_in[2];
    const float* b1    = (const float*)d_in[3];
    const float* W2    = (const float*)d_in[4];
    const float* b2    = (const float*)d_in[5];
    const float* Wc    = (const float*)d_in[6];
    const float* bc    = (const float*)d_in[7];
    float* out = (float*)d_out;

    float* ws      = (float*)d_ws;
    float* nrm_src = ws;                              // [R, N]
    float* nrm_dst = nrm_src + (size_t)R_REL * N_NODES; // [R, N]
    float* hw      = nrm_dst + (size_t)R_REL * N_NODES; // [N, 128]
    float* h1      = hw + (size_t)N_NODES * D_FEAT;     // [N, 128]
    float* h2      = h1 + (size_t)N_NODES * D_FEAT;     // [N, 128]

    const size_t featBytes = (size_t)N_NODES * D_FEAT * sizeof(float);
    const int gemmGrid   = N_NODES / 16;                       // 6250 exact
    const int scatGrid   = (E_EDGES * 32) / 256;               // 50000 exact
    const int elemGrid   = (N_NODES * D_FEAT + 255) / 256;     // 50000
    const int degGrid    = (R_REL * E_EDGES + 255) / 256;
    const int normGrid   = (2 * R_REL * N_NODES + 255) / 256;

    // degrees -> rsqrt norms
    hipMemsetAsync(nrm_src, 0, 2 * (size_t)R_REL * N_NODES * sizeof(float), stream);
    rgcn_deg_kernel<<<degGrid, 256, 0, stream>>>(edges, nrm_src, nrm_dst);
    rgcn_norm_kernel<<<normGrid, 256, 0, stream>>>(nrm_src, 2 * R_REL * N_NODES);

    // layer 1: h1 = relu(mean_r(scatter((x.*sn_r)@W1_r) .* dn_r + b1_r))
    hipMemsetAsync(h1, 0, featBytes, stream);
    for (int r = 0; r < R_REL; ++r) {
        const int* src = edges + (size_t)r * 2 * E_EDGES;
        const int* dst = src + E_EDGES;
        rgcn_gemm128_wmma<<<gemmGrid, 256, 0, stream>>>(
            x, nrm_src + (size_t)r * N_NODES,
            W1 + (size_t)r * D_FEAT * D_FEAT, hw);
        rgcn_scatter_kernel<<<scatGrid, 256, 0, stream>>>(
            hw, src, dst, nrm_dst + (size_t)r * N_NODES, h1);
    }
    rgcn_finalize_kernel<<<elemGrid, 256, 0, stream>>>(h1, b1, 1);

    // layer 2: h2 = mean_r(scatter((h1.*sn_r)@W2_r) .* dn_r + b2_r)
    hipMemsetAsync(h2, 0, featBytes, stream);
    for (int r = 0; r < R_REL; ++r) {
        const int* src = edges + (size_t)r * 2 * E_EDGES;
        const int* dst = src + E_EDGES;
        rgcn_gemm128_wmma<<<gemmGrid, 256, 0, stream>>>(
            h1, nrm_src + (size_t)r * N_NODES,
            W2 + (size_t)r * D_FEAT * D_FEAT, hw);
        rgcn_scatter_kernel<<<scatGrid, 256, 0, stream>>>(
            hw, src, dst, nrm_dst + (size_t)r * N_NODES, h2);
    }
    rgcn_finalize_kernel<<<elemGrid, 256, 0, stream>>>(h2, b2, 0);

    // classifier
    rgcn_cls_wmma<<<gemmGrid, 96, 0, stream>>>(h2, Wc, bc, out);
}